// Breadth_62088047230980
// MI455X (gfx1250) — compile-verified
//
#include <hip/hip_runtime.h>
#include <hip/hip_bf16.h>

typedef __attribute__((ext_vector_type(2))) float v2f;
typedef __attribute__((ext_vector_type(8))) float v8f;

#define IN_DIM 128
#define OUT_DIM 64
#define NEG_SLOPE 0.2f

// -------------------------------------------------------------------------
// K0: init accumulators.  out[N*64]=0, denom[N]=0, m_enc[N]=enc(-inf)
// enc(f): order-preserving map float->uint:  neg -> ~bits, pos -> bits|0x80000000
// enc(-inf) = ~0xFF800000 = 0x007FFFFF
// -------------------------------------------------------------------------
__global__ void gat_init(float* __restrict__ out, float* __restrict__ denom,
                         unsigned* __restrict__ m_enc, int n_nodes) {
    long long i = (long long)blockIdx.x * blockDim.x + threadIdx.x;
    long long total = (long long)n_nodes * OUT_DIM;
    if (i < total) out[i] = 0.0f;
    if (i < n_nodes) {
        denom[i] = 0.0f;
        m_enc[i] = 0x007FFFFFu;
    }
}

// -------------------------------------------------------------------------
// K1: feat = x @ W  using V_WMMA_F32_16X16X4_F32.
// One wave handles a 16-row M tile and all 4 N-subtiles (OUT_DIM=64).
// W is staged transposed in LDS: Wt[col][k], so B loads are ds_load_b64.
// A layout (16x4 f32): lanes 0-15: M=lane, {K=k,k+1}; lanes 16-31: M=lane-16, {K=k+2,k+3}
// B layout (4x16 f32): lanes 0-15: N=lane, {K=k,k+1}; lanes 16-31: N=lane-16, {K=k+2,k+3}
// C/D (16x16 f32): VGPR r: lanes 0-15 -> (M=r, N=lane); lanes 16-31 -> (M=r+8, N=lane-16)
// -------------------------------------------------------------------------
__global__ __launch_bounds__(256) void gat_gemm(const float* __restrict__ x,
                                                const float* __restrict__ W,
                                                float* __restrict__ feat,
                                                int n_nodes) {
    __shared__ float Wt[OUT_DIM][IN_DIM];   // 32 KB
    for (int i = threadIdx.x; i < IN_DIM * OUT_DIM; i += blockDim.x) {
        int k = i >> 6;     // row of W   (IN_DIM index)
        int c = i & 63;     // col of W   (OUT_DIM index)
        Wt[c][k] = W[i];
    }
    __syncthreads();

    const int wave = threadIdx.x >> 5;
    const int lane = threadIdx.x & 31;
    const int tile = blockIdx.x * (blockDim.x >> 5) + wave;   // 16-row tile index
    const int n_tiles = (n_nodes + 15) >> 4;
    if (tile >= n_tiles) return;                               // wave-uniform

    const int row0 = tile << 4;
    const int l16  = lane & 15;
    const int hi   = lane >> 4;            // 0: K pair (k,k+1), 1: (k+2,k+3)

    v8f acc0 = {}, acc1 = {}, acc2 = {}, acc3 = {};
    const float* xrow = x + (size_t)(row0 + l16) * IN_DIM;

    for (int k = 0; k < IN_DIM; k += 4) {
        v2f a = *(const v2f*)(xrow + k + 2 * hi);
        v2f b0 = *(const v2f*)&Wt[ 0 + l16][k + 2 * hi];
        v2f b1 = *(const v2f*)&Wt[16 + l16][k + 2 * hi];
        v2f b2 = *(const v2f*)&Wt[32 + l16][k + 2 * hi];
        v2f b3 = *(const v2f*)&Wt[48 + l16][k + 2 * hi];
        acc0 = __builtin_amdgcn_wmma_f32_16x16x4_f32(false, a, false, b0, (short)0, acc0, false, false);
        acc1 = __builtin_amdgcn_wmma_f32_16x16x4_f32(false, a, false, b1, (short)0, acc1, false, false);
        acc2 = __builtin_amdgcn_wmma_f32_16x16x4_f32(false, a, false, b2, (short)0, acc2, false, false);
        acc3 = __builtin_amdgcn_wmma_f32_16x16x4_f32(false, a, false, b3, (short)0, acc3, false, false);
    }

    // store 16x64 tile; lane holds column (nsub*16 + l16) of rows hi*8 + 0..7
    float* o = feat + ((size_t)(row0 + hi * 8)) * OUT_DIM + l16;
#pragma unroll
    for (int r = 0; r < 8; ++r) {
        o[(size_t)r * OUT_DIM +  0] = acc0[r];
        o[(size_t)r * OUT_DIM + 16] = acc1[r];
        o[(size_t)r * OUT_DIM + 32] = acc2[r];
        o[(size_t)r * OUT_DIM + 48] = acc3[r];
    }
}

// -------------------------------------------------------------------------
// K2: el[i] = feat[i,:] . attn_l ; er[i] = feat[i,:] . attn_r   (wave per node)
// -------------------------------------------------------------------------
__global__ void gat_logits(const float* __restrict__ feat,
                           const float* __restrict__ attn_l,
                           const float* __restrict__ attn_r,
                           float* __restrict__ el, float* __restrict__ er,
                           int n_nodes) {
    const int lane  = threadIdx.x & 31;
    const int waveg = (blockIdx.x * blockDim.x + threadIdx.x) >> 5;
    const int nwav  = (gridDim.x * blockDim.x) >> 5;
    const float al0 = attn_l[lane], al1 = attn_l[lane + 32];
    const float ar0 = attn_r[lane], ar1 = attn_r[lane + 32];
    for (int node = waveg; node < n_nodes; node += nwav) {
        const float* f = feat + (size_t)node * OUT_DIM;
        float f0 = f[lane], f1 = f[lane + 32];
        float pl = f0 * al0 + f1 * al1;
        float pr = f0 * ar0 + f1 * ar1;
#pragma unroll
        for (int off = 16; off; off >>= 1) {
            pl += __shfl_xor(pl, off, 32);
            pr += __shfl_xor(pr, off, 32);
        }
        if (lane == 0) { el[node] = pl; er[node] = pr; }
    }
}

// -------------------------------------------------------------------------
// K3: e = leaky_relu(el[src]+er[dst]); segment max via atomicMax on encoded uint
// -------------------------------------------------------------------------
__global__ void gat_edge_score(const int* __restrict__ src, const int* __restrict__ dst,
                               const float* __restrict__ el, const float* __restrict__ er,
                               float* __restrict__ e, unsigned* __restrict__ m_enc,
                               int n_edges) {
    int i = blockIdx.x * blockDim.x + threadIdx.x;
    if (i >= n_edges) return;
    int d = dst[i];
    float v = el[src[i]] + er[d];
    v = v > 0.0f ? v : NEG_SLOPE * v;
    e[i] = v;
    unsigned bits = __float_as_uint(v);
    unsigned enc  = (bits & 0x80000000u) ? ~bits : (bits | 0x80000000u);
    atomicMax(&m_enc[d], enc);
}

// -------------------------------------------------------------------------
// K4: ex = exp(e - m[dst]) (in place), denom[dst] += ex
// -------------------------------------------------------------------------
__global__ void gat_exp(const int* __restrict__ dst, const unsigned* __restrict__ m_enc,
                        float* __restrict__ e, float* __restrict__ denom, int n_edges) {
    int i = blockIdx.x * blockDim.x + threadIdx.x;
    if (i >= n_edges) return;
    int d = dst[i];
    unsigned enc  = m_enc[d];
    unsigned bits = (enc & 0x80000000u) ? (enc & 0x7FFFFFFFu) : ~enc;
    float m  = __uint_as_float(bits);
    float ex = expf(e[i] - m);
    e[i] = ex;
    atomicAdd(&denom[d], ex);
}

// -------------------------------------------------------------------------
// K5: out[dst,:] += (ex/denom[dst]) * feat[src,:]
// thread-per-(edge, col): fully coalesced gather + coalesced L2 atomics
// (feat and out both fit in MI455X's 192 MB L2)
// -------------------------------------------------------------------------
__global__ void gat_aggregate(const int* __restrict__ src, const int* __restrict__ dst,
                              const float* __restrict__ ex, const float* __restrict__ denom,
                              const float* __restrict__ feat, float* __restrict__ out,
                              int n_edges) {
    long long t = (long long)blockIdx.x * blockDim.x + threadIdx.x;
    int edge = (int)(t >> 6);
    if (edge >= n_edges) return;
    int c = (int)t & 63;
    int d = dst[edge], s = src[edge];
    float alpha = ex[edge] / denom[d];
    atomicAdd(&out[(size_t)d * OUT_DIM + c], alpha * feat[(size_t)s * OUT_DIM + c]);
}

// -------------------------------------------------------------------------
// K6: out = tanh(out + bias)
// -------------------------------------------------------------------------
__global__ void gat_finalize(float* __restrict__ out, const float* __restrict__ bias,
                             int n_nodes) {
    long long i = (long long)blockIdx.x * blockDim.x + threadIdx.x;
    long long total = (long long)n_nodes * OUT_DIM;
    if (i >= total) return;
    out[i] = tanhf(out[i] + bias[i & 63]);
}

static inline size_t align256(size_t x) { return (x + 255) & ~(size_t)255; }

extern "C" void kernel_launch(void* const* d_in, const int* in_sizes, int n_in,
                              void* d_out, int out_size, void* d_ws, size_t ws_size,
                              hipStream_t stream) {
    const float* x      = (const float*)d_in[0];
    const int*   src    = (const int*)d_in[1];
    const int*   dst    = (const int*)d_in[2];
    const float* W      = (const float*)d_in[3];
    const float* attn_l = (const float*)d_in[4];
    const float* attn_r = (const float*)d_in[5];
    const float* bias   = (const float*)d_in[6];
    float*       out    = (float*)d_out;

    const int N = in_sizes[0] / IN_DIM;
    const int E = in_sizes[1];

    char* ws = (char*)d_ws;
    float*    feat  = (float*)ws;    ws += align256((size_t)N * OUT_DIM * sizeof(float));
    float*    e     = (float*)ws;    ws += align256((size_t)E * sizeof(float));
    float*    el    = (float*)ws;    ws += align256((size_t)N * sizeof(float));
    float*    er    = (float*)ws;    ws += align256((size_t)N * sizeof(float));
    float*    denom = (float*)ws;    ws += align256((size_t)N * sizeof(float));
    unsigned* m_enc = (unsigned*)ws; ws += align256((size_t)N * sizeof(unsigned));

    const int BLK = 256;

    // K0: init (covers N*64 elements)
    {
        long long total = (long long)N * OUT_DIM;
        int blocks = (int)((total + BLK - 1) / BLK);
        gat_init<<<blocks, BLK, 0, stream>>>(out, denom, m_enc, N);
    }
    // K1: WMMA GEMM
    {
        int n_tiles = (N + 15) / 16;          // one wave per tile
        int waves_per_block = BLK / 32;
        int blocks = (n_tiles + waves_per_block - 1) / waves_per_block;
        gat_gemm<<<blocks, BLK, 0, stream>>>(x, W, feat, N);
    }
    // K2: attention logits
    {
        int blocks = 1024;
        gat_logits<<<blocks, BLK, 0, stream>>>(feat, attn_l, attn_r, el, er, N);
    }
    // K3: edge scores + segment max
    {
        int blocks = (E + BLK - 1) / BLK;
        gat_edge_score<<<blocks, BLK, 0, stream>>>(src, dst, el, er, e, m_enc, E);
    }
    // K4: exp + segment sum
    {
        int blocks = (E + BLK - 1) / BLK;
        gat_exp<<<blocks, BLK, 0, stream>>>(dst, m_enc, e, denom, E);
    }
    // K5: weighted scatter-aggregate
    {
        long long total = (long long)E * OUT_DIM;
        int blocks = (int)((total + BLK - 1) / BLK);
        gat_aggregate<<<blocks, BLK, 0, stream>>>(src, dst, e, denom, feat, out, E);
    }
    // K6: bias + tanh
    {
        long long total = (long long)N * OUT_DIM;
        int blocks = (int)((total + BLK - 1) / BLK);
        gat_finalize<<<blocks, BLK, 0, stream>>>(out, bias, N);
    }
}